// KDPointTrans_ablation_cp_78898549228174
// MI455X (gfx1250) — compile-verified
//
#include <hip/hip_runtime.h>
#include <stdint.h>

#define B_    16
#define C_    6
#define H_    512
#define W_    512
#define HW_   (H_ * W_)          // 262144
#define K_    500
#define CK_   (C_ * K_)          // 3000
#define NP_   (B_ * C_)          // 96 planes
#define SPLIT 8                  // chunks per plane -> 768 streaming WGs
#define CHUNK (HW_ / SPLIT)      // 32768 elements
#define CHUNKV (CHUNK / 4)       // 8192 float4
#define NBINS  4096
#define DEFCAP 512
#define BNDCAP 2048
#define P2     4096              // batch-merge bitonic size

typedef float  __attribute__((ext_vector_type(4))) fvec4;

// ---- workspace layout (uint32 words) ----
#define WS_HIST   0                                   // NP_*NBINS
#define WS_DCNT   (WS_HIST + NP_ * NBINS)             // NP_
#define WS_BCNT   (WS_DCNT + NP_)                     // NP_
#define WS_ZEND   (WS_BCNT + NP_)                     // zeroed prefix
#define WS_DEFK   (WS_ZEND)                           // NP_*DEFCAP
#define WS_DEFI   (WS_DEFK + NP_ * DEFCAP)
#define WS_BNDK   (WS_DEFI + NP_ * DEFCAP)            // NP_*BNDCAP
#define WS_BNDI   (WS_BNDK + NP_ * BNDCAP)
#define WS_KEYS   (WS_BNDI + NP_ * BNDCAP)            // NP_*K_
#define WS_PACK   (WS_KEYS + NP_ * K_)

// ---- order-preserving float<->uint key (monotone with sigmoid) ----
__device__ __forceinline__ uint32_t orderKey(float f) {
  uint32_t u = __float_as_uint(f);
  return u ^ (uint32_t)(((int32_t)u >> 31) | 0x80000000u);
}
__device__ __forceinline__ float unorderKey(uint32_t k) {
  uint32_t u = (k & 0x80000000u) ? (k ^ 0x80000000u) : ~k;
  return __uint_as_float(u);
}

// =====================================================================
// Kernel Z: zero histograms + counters (graph-safe, deterministic)
// =====================================================================
extern "C" __global__ __launch_bounds__(256)
void cp_zero(uint32_t* __restrict__ ws) {
  int i = blockIdx.x * 256 + threadIdx.x;
  if (i < WS_ZEND) ws[i] = 0u;
}

// =====================================================================
// Kernel A: chunked streaming histogram (HBM-bound pass)
// grid = NP_*SPLIT, block = 256
// =====================================================================
extern "C" __global__ __launch_bounds__(256)
void cp_hist(const float* __restrict__ hm, uint32_t* __restrict__ ws) {
  __shared__ uint32_t lh[NBINS];
  const int wg  = blockIdx.x;
  const int bc  = wg / SPLIT;
  const int s   = wg % SPLIT;
  const int tid = threadIdx.x;

  for (int i = tid; i < NBINS; i += 256) lh[i] = 0u;
  __syncthreads();

  const fvec4* p = (const fvec4*)hm + (size_t)bc * (HW_ / 4) + (size_t)s * CHUNKV;
  for (int i = tid; i < CHUNKV; i += 256) {
    __builtin_prefetch(p + i + 2048, 0, 0);          // global_prefetch_b8
    fvec4 v = p[i];
    atomicAdd(&lh[orderKey(v.x) >> 20], 1u);
    atomicAdd(&lh[orderKey(v.y) >> 20], 1u);
    atomicAdd(&lh[orderKey(v.z) >> 20], 1u);
    atomicAdd(&lh[orderKey(v.w) >> 20], 1u);
  }
  __syncthreads();

  uint32_t* gh = ws + WS_HIST + (size_t)bc * NBINS;
  for (int i = tid; i < NBINS; i += 256) {
    uint32_t n = lh[i];
    if (n) atomicAdd(&gh[i], n);
  }
}

// =====================================================================
// Kernel C: per-chunk threshold (suffix scan of plane histogram) + compact
// grid = NP_*SPLIT, block = 256. Re-read is L2-resident (100MB < 192MB L2),
// marked non-temporal (last use).
// =====================================================================
extern "C" __global__ __launch_bounds__(256)
void cp_compact(const float* __restrict__ hm, uint32_t* __restrict__ ws) {
  __shared__ uint32_t suf[NBINS];
  __shared__ uint32_t sThr;
  const int wg  = blockIdx.x;
  const int bc  = wg / SPLIT;
  const int s   = wg % SPLIT;
  const int tid = threadIdx.x;

  const uint32_t* gh = ws + WS_HIST + (size_t)bc * NBINS;
  for (int i = tid; i < NBINS; i += 256) suf[i] = gh[i];
  __syncthreads();

  // Hillis-Steele suffix sums: suf[t] = count(bin >= t)
  uint32_t tmp[NBINS / 256];
  for (uint32_t off = 1; off < NBINS; off <<= 1) {
#pragma unroll
    for (int k = 0; k < NBINS / 256; ++k) {
      int i = tid + k * 256;
      uint32_t v = suf[i];
      if (i + (int)off < NBINS) v += suf[i + off];
      tmp[k] = v;
    }
    __syncthreads();
#pragma unroll
    for (int k = 0; k < NBINS / 256; ++k) suf[tid + k * 256] = tmp[k];
    __syncthreads();
  }

  // threshold t: suf[t] >= K and (t == last or suf[t+1] < K)
#pragma unroll
  for (int k = 0; k < NBINS / 256; ++k) {
    int t = tid + k * 256;
    uint32_t here = suf[t];
    uint32_t next = (t + 1 < NBINS) ? suf[t + 1] : 0u;
    if (here >= K_ && next < K_) sThr = (uint32_t)t;
  }
  __syncthreads();
  const uint32_t thr = sThr;

  uint32_t* dCnt = ws + WS_DCNT;
  uint32_t* bCnt = ws + WS_BCNT;
  uint32_t* defK = ws + WS_DEFK + (size_t)bc * DEFCAP;
  uint32_t* defI = ws + WS_DEFI + (size_t)bc * DEFCAP;
  uint32_t* bndK = ws + WS_BNDK + (size_t)bc * BNDCAP;
  uint32_t* bndI = ws + WS_BNDI + (size_t)bc * BNDCAP;

  const fvec4* p = (const fvec4*)hm + (size_t)bc * (HW_ / 4) + (size_t)s * CHUNKV;
  for (int i = tid; i < CHUNKV; i += 256) {
    fvec4 v = __builtin_nontemporal_load(p + i);     // last-use streaming read
    uint32_t ks[4] = {orderKey(v.x), orderKey(v.y), orderKey(v.z), orderKey(v.w)};
#pragma unroll
    for (int j = 0; j < 4; ++j) {
      uint32_t bin = ks[j] >> 20;
      if (bin > thr) {
        uint32_t pz = atomicAdd(&dCnt[bc], 1u);
        if (pz < DEFCAP) { defK[pz] = ks[j]; defI[pz] = (uint32_t)(s * CHUNK + i * 4 + j); }
      } else if (bin == thr) {
        uint32_t pz = atomicAdd(&bCnt[bc], 1u);
        if (pz < BNDCAP) { bndK[pz] = ks[j]; bndI[pz] = (uint32_t)(s * CHUNK + i * 4 + j); }
      }
    }
  }
}

// =====================================================================
// Kernel D: per-plane boundary sort + emit exactly K entries
// grid = NP_, block = 512. Boundary list pulled via CDNA5 async copy engine.
// =====================================================================
extern "C" __global__ __launch_bounds__(512)
void cp_emit(uint32_t* __restrict__ ws) {
  __shared__ uint32_t bK[BNDCAP], bI[BNDCAP];
  const int bc  = blockIdx.x;
  const int c   = bc % C_;
  const int tid = threadIdx.x;

  uint32_t nd = ws[WS_DCNT + bc]; if (nd > DEFCAP) nd = DEFCAP;  // < K by construction
  uint32_t bn = ws[WS_BCNT + bc]; if (bn > BNDCAP) bn = BNDCAP;

  const uint32_t* gbK = ws + WS_BNDK + (size_t)bc * BNDCAP;
  const uint32_t* gbI = ws + WS_BNDI + (size_t)bc * BNDCAP;

  // async global->LDS (ASYNCcnt), EXEC-predicated so only valid entries move
  for (int i = tid; i < BNDCAP; i += 512) {
    if ((uint32_t)i < bn) {
      uint32_t lk = (uint32_t)(uintptr_t)&bK[i];
      uint32_t li = (uint32_t)(uintptr_t)&bI[i];
      const uint32_t* ak = gbK + i;
      const uint32_t* ai = gbI + i;
      asm volatile("global_load_async_to_lds_b32 %0, %1, off" :: "v"(lk), "v"(ak) : "memory");
      asm volatile("global_load_async_to_lds_b32 %0, %1, off" :: "v"(li), "v"(ai) : "memory");
    } else {
      bK[i] = 0u; bI[i] = 0u;
    }
  }
  asm volatile("s_wait_asynccnt 0" ::: "memory");
  __syncthreads();

  // bitonic sort boundary bin, descending by key
  for (unsigned k = 2; k <= BNDCAP; k <<= 1) {
    for (unsigned j = k >> 1; j > 0; j >>= 1) {
      for (unsigned i = tid; i < BNDCAP; i += 512) {
        unsigned ixj = i ^ j;
        if (ixj > i) {
          bool desc = ((i & k) == 0);
          uint32_t a = bK[i], b = bK[ixj];
          bool sw = desc ? (a < b) : (a > b);
          if (sw) {
            bK[i] = b; bK[ixj] = a;
            uint32_t t2 = bI[i]; bI[i] = bI[ixj]; bI[ixj] = t2;
          }
        }
      }
      __syncthreads();
    }
  }

  const uint32_t* dK = ws + WS_DEFK + (size_t)bc * DEFCAP;
  const uint32_t* dI = ws + WS_DEFI + (size_t)bc * DEFCAP;
  uint32_t* outK = ws + WS_KEYS + (size_t)bc * K_;
  uint32_t* outP = ws + WS_PACK + (size_t)bc * K_;
  for (int m = tid; m < K_; m += 512) {
    uint32_t key, idx;
    if ((uint32_t)m < nd)                 { key = dK[m]; idx = dI[m]; }
    else if ((uint32_t)m - nd < BNDCAP)   { uint32_t r = (uint32_t)m - nd; key = bK[r]; idx = bI[r]; }
    else                                  { key = 0u; idx = 0u; }
    outK[m] = key;
    outP[m] = ((uint32_t)c << 20) | idx;              // idx < 2^18, class < 8
  }
}

// =====================================================================
// Kernel E: per-batch merge of C*K candidates + box decode + outputs
// grid = B_, block = 1024
// =====================================================================
extern "C" __global__ __launch_bounds__(1024)
void cp_merge_decode(const uint32_t* __restrict__ ws,
                     const float* __restrict__ center,
                     const float* __restrict__ center_z,
                     const float* __restrict__ dim_feat,
                     const float* __restrict__ rot,
                     const float* __restrict__ vel,
                     float* __restrict__ out) {
  __shared__ uint32_t sK[P2];
  __shared__ uint32_t sP[P2];
  const int b   = blockIdx.x;
  const int tid = threadIdx.x;

  const uint32_t* gk = ws + WS_KEYS + (size_t)b * CK_;
  const uint32_t* gp = ws + WS_PACK + (size_t)b * CK_;
  for (int i = tid; i < CK_; i += 1024) {
    uint32_t lk = (uint32_t)(uintptr_t)&sK[i];
    uint32_t lp = (uint32_t)(uintptr_t)&sP[i];
    const uint32_t* ak = gk + i;
    const uint32_t* ap = gp + i;
    asm volatile("global_load_async_to_lds_b32 %0, %1, off" :: "v"(lk), "v"(ak) : "memory");
    asm volatile("global_load_async_to_lds_b32 %0, %1, off" :: "v"(lp), "v"(ap) : "memory");
  }
  for (int i = CK_ + tid; i < P2; i += 1024) { sK[i] = 0u; sP[i] = 0u; }
  asm volatile("s_wait_asynccnt 0" ::: "memory");
  __syncthreads();

  // bitonic sort 4096 descending by key
  for (unsigned k = 2; k <= P2; k <<= 1) {
    for (unsigned j = k >> 1; j > 0; j >>= 1) {
      for (unsigned i = tid; i < P2; i += 1024) {
        unsigned ixj = i ^ j;
        if (ixj > i) {
          bool desc = ((i & k) == 0);
          uint32_t a = sK[i], bb = sK[ixj];
          bool sw = desc ? (a < bb) : (a > bb);
          if (sw) {
            sK[i] = bb; sK[ixj] = a;
            uint32_t t2 = sP[i]; sP[i] = sP[ixj]; sP[ixj] = t2;
          }
        }
      }
      __syncthreads();
    }
  }

  if (tid < K_) {
    uint32_t key = sK[tid], pk = sP[tid];
    int cls = (int)(pk >> 20);
    int ind = (int)(pk & 0xFFFFFu);
    float logit = unorderKey(key);
    float score = 1.0f / (1.0f + __expf(-logit));

    float xs = (float)(ind % W_);
    float ys = (float)(ind / W_);
    size_t o = (size_t)ind;

    float cx = center[((size_t)b * 2 + 0) * HW_ + o];
    float cy = center[((size_t)b * 2 + 1) * HW_ + o];
    float cz = center_z[(size_t)b * HW_ + o];
    float d0 = __expf(dim_feat[((size_t)b * 3 + 0) * HW_ + o]);
    float d1 = __expf(dim_feat[((size_t)b * 3 + 1) * HW_ + o]);
    float d2 = __expf(dim_feat[((size_t)b * 3 + 2) * HW_ + o]);
    float rc = rot[((size_t)b * 2 + 0) * HW_ + o];
    float rs = rot[((size_t)b * 2 + 1) * HW_ + o];
    float ang = atan2f(rs, rc);
    float v0 = vel[((size_t)b * 2 + 0) * HW_ + o];
    float v1 = vel[((size_t)b * 2 + 1) * HW_ + o];

    float X = (xs + cx) * 0.2f - 51.2f;   // STRIDE*VOXEL + PCR[0]
    float Y = (ys + cy) * 0.2f - 51.2f;

    bool m = (X >= -61.2f) && (X <= 61.2f) &&
             (Y >= -61.2f) && (Y <= 61.2f) &&
             (cz >= -10.0f) && (cz <= 10.0f) &&
             (score > 0.1f);

    float* box = out + ((size_t)b * K_ + tid) * 9;
    box[0] = X; box[1] = Y; box[2] = cz;
    box[3] = d0; box[4] = d1; box[5] = d2;
    box[6] = ang; box[7] = v0; box[8] = v1;

    const size_t BK = (size_t)B_ * K_;
    out[BK * 9  + (size_t)b * K_ + tid] = score;           // final_scores
    out[BK * 10 + (size_t)b * K_ + tid] = (float)cls;      // final_class_ids
    out[BK * 11 + (size_t)b * K_ + tid] = (float)ind;      // inds
    out[BK * 12 + (size_t)b * K_ + tid] = m ? 1.0f : 0.0f; // mask
  }
}

extern "C" void kernel_launch(void* const* d_in, const int* in_sizes, int n_in,
                              void* d_out, int out_size, void* d_ws, size_t ws_size,
                              hipStream_t stream) {
  (void)in_sizes; (void)n_in; (void)out_size; (void)ws_size;
  const float* heatmap  = (const float*)d_in[0];
  const float* center   = (const float*)d_in[1];
  const float* center_z = (const float*)d_in[2];
  const float* dim_feat = (const float*)d_in[3];
  const float* rot      = (const float*)d_in[4];
  const float* vel      = (const float*)d_in[5];
  // d_in[6] is K on-device; fixed at 500 by the reference setup.

  uint32_t* ws = (uint32_t*)d_ws;

  cp_zero<<<(WS_ZEND + 255) / 256, 256, 0, stream>>>(ws);
  cp_hist<<<NP_ * SPLIT, 256, 0, stream>>>(heatmap, ws);
  cp_compact<<<NP_ * SPLIT, 256, 0, stream>>>(heatmap, ws);
  cp_emit<<<NP_, 512, 0, stream>>>(ws);
  cp_merge_decode<<<B_, 1024, 0, stream>>>(ws, center, center_z, dim_feat,
                                           rot, vel, (float*)d_out);
}